// BahdanauScorer_64381559767051
// MI455X (gfx1250) — compile-verified
//
#include <hip/hip_runtime.h>
#include <hip/hip_bf16.h>

// Shapes from the reference
#define SRC_LEN 256
#define TRG_LEN 256
#define BATCH   32
#define HID     512
#define ATT     128

typedef __attribute__((ext_vector_type(2))) float v2f;
typedef __attribute__((ext_vector_type(8))) float v8f;

__device__ __forceinline__ float fast_tanh(float x) {
#if __has_builtin(__builtin_amdgcn_tanhf)
    return __builtin_amdgcn_tanhf(x);   // gfx1250 v_tanh_f32
#else
    return tanhf(x);
#endif
}

// ---------------------------------------------------------------------------
// Kernel 1: fp32 WMMA projection  out[M,128] = X[M,512] * W[128,512]^T (+bias)
// One wave computes one 16x16 output tile, stepping K by 4 with
// v_wmma_f32_16x16x4_f32 (128 WMMA ops per tile).
//
// Per-lane layout (ISA 7.12.2, wave32):
//   A 16x4 : lane%16 = M row;  VGPR i holds K = k0 + 2*(lane/16) + i
//   B 4x16 : lane%16 = N col;  VGPR i holds K = k0 + 2*(lane/16) + i
//   C 16x16: VGPR i -> row (i + 8*(lane/16)), col lane%16
// Both A and B therefore read a contiguous float2 along the hid dimension.
// ---------------------------------------------------------------------------
__global__ __launch_bounds__(256) void proj_wmma_f32(
    const float* __restrict__ X,     // [M, HID]  (row = s*BATCH + b, contiguous)
    const float* __restrict__ W,     // [ATT, HID]
    const float* __restrict__ bias,  // [ATT] or nullptr
    float* __restrict__ out,         // [M, ATT]
    int M)
{
    const int wave   = threadIdx.x >> 5;
    const int lane   = threadIdx.x & 31;
    const int half   = lane >> 4;          // 0: lanes 0-15, 1: lanes 16-31
    const int l16    = lane & 15;

    const int nTilesN = ATT / 16;          // 8
    const int tileId  = blockIdx.x * 8 + wave;
    const int nTiles  = (M / 16) * nTilesN;
    if (tileId >= nTiles) return;          // uniform per wave

    const int mt = tileId / nTilesN;
    const int nt = tileId % nTilesN;

    const float* __restrict__ arow = X + (size_t)(mt * 16 + l16) * HID + 2 * half;
    const float* __restrict__ brow = W + (size_t)(nt * 16 + l16) * HID + 2 * half;

    v8f c = {};
#pragma unroll 8
    for (int k = 0; k < HID; k += 4) {
        v2f a = *(const v2f*)(arow + k);
        v2f b = *(const v2f*)(brow + k);
        // (neg_a, A, neg_b, B, c_mod, C, reuse_a, reuse_b)
        c = __builtin_amdgcn_wmma_f32_16x16x4_f32(
                false, a, false, b, (short)0, c, false, false);
    }

    const float bval = bias ? bias[nt * 16 + l16] : 0.0f;
    float* __restrict__ orow = out + (size_t)(mt * 16 + 8 * half) * ATT + nt * 16 + l16;
#pragma unroll
    for (int i = 0; i < 8; ++i) {
        orow[(size_t)i * ATT] = c[i] + bval;
    }
}

// ---------------------------------------------------------------------------
// Kernel 2: scores[t,b,s] = sum_a v[a] * tanh(enc_att[s,b,a] + dec_att[t,b,a])
// Block: fixed b, 32t x 32s tile. dec/enc tiles + v_a staged in LDS
// (32*128*4*2 + 512 = ~32.5 KB). Each thread: 1 t, 4 s. Inner loop over a in
// float4 steps -> 16 v_tanh_f32 + 16 fma per iteration; se/sv reads are
// wave-broadcasts (conflict free); sd is read one row per lane (row tl, 128-float stride).
// ---------------------------------------------------------------------------
__global__ __launch_bounds__(256) void bahdanau_score(
    const float* __restrict__ enc_att,  // [SRC, BATCH, ATT]
    const float* __restrict__ dec_att,  // [TRG, BATCH, ATT]
    const float* __restrict__ v_a,      // [ATT]
    float* __restrict__ out)            // [TRG, BATCH, SRC]
{
    __shared__ float sd[32][ATT];  // dec rows (t)
    __shared__ float se[32][ATT];  // enc rows (s)
    __shared__ float sv[ATT];

    const int b   = blockIdx.z;
    const int t0  = blockIdx.y * 32;
    const int s0  = blockIdx.x * 32;
    const int tid = threadIdx.x;

    // Cooperative tile load: 32 rows x 128 floats each = 1024 float4 per matrix
    for (int i = tid; i < 32 * (ATT / 4); i += 256) {
        const int row = i / (ATT / 4);
        const int col = (i % (ATT / 4)) * 4;
        *(float4*)&sd[row][col] =
            *(const float4*)&dec_att[((size_t)(t0 + row) * BATCH + b) * ATT + col];
        *(float4*)&se[row][col] =
            *(const float4*)&enc_att[((size_t)(s0 + row) * BATCH + b) * ATT + col];
    }
    if (tid < ATT) sv[tid] = v_a[tid];
    __syncthreads();

    const int tl = tid & 31;   // local t
    const int sg = tid >> 5;   // 0..7 -> s = sg + 8*j

    float acc0 = 0.f, acc1 = 0.f, acc2 = 0.f, acc3 = 0.f;
#pragma unroll 4
    for (int a = 0; a < ATT; a += 4) {
        const float4 d = *(const float4*)&sd[tl][a];
        const float4 v = *(const float4*)&sv[a];

        float4 e;
        e = *(const float4*)&se[sg + 0][a];
        acc0 += v.x * fast_tanh(d.x + e.x);
        acc0 += v.y * fast_tanh(d.y + e.y);
        acc0 += v.z * fast_tanh(d.z + e.z);
        acc0 += v.w * fast_tanh(d.w + e.w);

        e = *(const float4*)&se[sg + 8][a];
        acc1 += v.x * fast_tanh(d.x + e.x);
        acc1 += v.y * fast_tanh(d.y + e.y);
        acc1 += v.z * fast_tanh(d.z + e.z);
        acc1 += v.w * fast_tanh(d.w + e.w);

        e = *(const float4*)&se[sg + 16][a];
        acc2 += v.x * fast_tanh(d.x + e.x);
        acc2 += v.y * fast_tanh(d.y + e.y);
        acc2 += v.z * fast_tanh(d.z + e.z);
        acc2 += v.w * fast_tanh(d.w + e.w);

        e = *(const float4*)&se[sg + 24][a];
        acc3 += v.x * fast_tanh(d.x + e.x);
        acc3 += v.y * fast_tanh(d.y + e.y);
        acc3 += v.z * fast_tanh(d.z + e.z);
        acc3 += v.w * fast_tanh(d.w + e.w);
    }

    const int t = t0 + tl;
    float* __restrict__ orow = out + ((size_t)t * BATCH + b) * SRC_LEN + s0;
    orow[sg +  0] = acc0;
    orow[sg +  8] = acc1;
    orow[sg + 16] = acc2;
    orow[sg + 24] = acc3;
}

// ---------------------------------------------------------------------------
extern "C" void kernel_launch(void* const* d_in, const int* in_sizes, int n_in,
                              void* d_out, int out_size, void* d_ws, size_t ws_size,
                              hipStream_t stream) {
    const float* dec_out  = (const float*)d_in[0];  // [TRG, BATCH, HID]
    const float* enc_outs = (const float*)d_in[1];  // [SRC, BATCH, HID]
    const float* W_s      = (const float*)d_in[2];  // [ATT, HID]
    const float* W_t      = (const float*)d_in[3];  // [ATT, HID]
    const float* b_t      = (const float*)d_in[4];  // [ATT]
    const float* v_a      = (const float*)d_in[5];  // [ATT, 1]

    float* scores = (float*)d_out;                  // [TRG, BATCH, SRC]

    const int M = SRC_LEN * BATCH;                  // 8192 rows per projection
    float* enc_att = (float*)d_ws;                          // M * ATT floats (4 MB)
    float* dec_att = enc_att + (size_t)M * ATT;             // M * ATT floats (4 MB)

    // Projections: 4096 16x16 tiles each, 8 waves/block -> 512 blocks
    const int tiles  = (M / 16) * (ATT / 16);
    const int blocks = (tiles + 7) / 8;
    proj_wmma_f32<<<blocks, 256, 0, stream>>>(enc_outs, W_s, nullptr, enc_att, M);
    proj_wmma_f32<<<blocks, 256, 0, stream>>>(dec_out,  W_t, b_t,     dec_att, M);

    // Score: 8 x 8 (s,t) tiles x 32 batches
    dim3 grid(SRC_LEN / 32, TRG_LEN / 32, BATCH);
    bahdanau_score<<<grid, 256, 0, stream>>>(enc_att, dec_att, v_a, scores);
}